// PostprocessingYoloV8Mask_69784628625814
// MI455X (gfx1250) — compile-verified
//
#include <hip/hip_runtime.h>
#include <stdint.h>

// ---------------- problem constants (from reference) ----------------
#define NB      8        // batch
#define NCH     116      // channels per anchor (4 box + 80 cls + 32 mask)
#define NA      8400     // anchors
#define NCLS    80
#define NMC     32       // mask coefficients
#define CONF    0.25f
#define IOU_THR 0.45f
#define MAX_DET 100
#define MAX_WH  7680.0f
#define PH      160
#define PW      160
#define PN      (PH*PW)  // 25600 proto pixels
#define OH      320
#define OW      320
#define MPAD    112      // MAX_DET padded to 7 WMMA 16-row tiles

typedef __attribute__((ext_vector_type(2))) float v2f;
typedef __attribute__((ext_vector_type(8))) float v8f;

// gfx1250 async global->LDS path (ASYNCcnt), guarded so either toolchain compiles.
// Probe (round 2 diagnostic): param type is 'int __attribute__((vector_size(16)))*'.
#if defined(__gfx1250__) && __has_builtin(__builtin_amdgcn_global_load_async_to_lds_b128) && __has_builtin(__builtin_amdgcn_s_wait_asynccnt)
#define USE_ASYNC_LDS 1
typedef int v4i_vs __attribute__((vector_size(16)));
#endif

// ---------------- workspace layout (in floats) ----------------
static constexpr size_t WS_S    = 0;                               // [NB][NA]        score or -1
static constexpr size_t WS_BOX  = WS_S    + (size_t)NB*NA;         // [NB][NA][4]     xyxy
static constexpr size_t WS_LAB  = WS_BOX  + (size_t)NB*NA*4;       // [NB][NA]        (int)
static constexpr size_t WS_ORD  = WS_LAB  + (size_t)NB*NA;         // [NB][NA]        (int) sorted order
static constexpr size_t WS_MC   = WS_ORD  + (size_t)NB*NA;         // [NB][MPAD][NMC] selected coefs (zero pad)
static constexpr size_t WS_M160 = WS_MC   + (size_t)NB*MPAD*NMC;   // [NB][MAX_DET][PH][PW] cropped sigmoid

// ================= kernel 1: decode anchors =================
__global__ void yolo_decode_kernel(const float* __restrict__ preds,
                                   float* __restrict__ s,
                                   float* __restrict__ box,
                                   int*   __restrict__ lab) {
    int g = blockIdx.x * blockDim.x + threadIdx.x;
    if (g >= NB * NA) return;
    int img = g / NA, a = g % NA;
    const float* p = preds + (size_t)img * NCH * NA;
    float cx = p[0*NA + a], cy = p[1*NA + a];
    float w  = p[2*NA + a], h  = p[3*NA + a];
    float best = -1e30f; int bi = 0;
    #pragma unroll 4
    for (int c = 0; c < NCLS; ++c) {
        float v = p[(size_t)(4 + c) * NA + a];
        if (v > best) { best = v; bi = c; }   // strict > keeps first occurrence (jnp.argmax)
    }
    bool valid = best > CONF;
    s[g] = valid ? best : -1.0f;
    box[(size_t)g*4 + 0] = cx - 0.5f*w;
    box[(size_t)g*4 + 1] = cy - 0.5f*h;
    box[(size_t)g*4 + 2] = cx + 0.5f*w;
    box[(size_t)g*4 + 3] = cy + 0.5f*h;
    lab[g] = bi;
}

// ================= kernel 2: stable descending rank (argsort(-s)) =================
__global__ void yolo_rank_kernel(const float* __restrict__ s, int* __restrict__ order) {
    __shared__ float tile[256];
    int img = blockIdx.y;
    int a = blockIdx.x * blockDim.x + threadIdx.x;
    const float* si = s + (size_t)img * NA;
    float mys = (a < NA) ? si[a] : 0.0f;
    int rank = 0;
    for (int base = 0; base < NA; base += 256) {
        int j = base + threadIdx.x;
        tile[threadIdx.x] = (j < NA) ? si[j] : -1e30f;
        __syncthreads();
        int lim = (NA - base < 256) ? (NA - base) : 256;
        if (a < NA) {
            for (int t = 0; t < lim; ++t) {
                float v = tile[t];
                rank += ((v > mys) || (v == mys && (base + t) < a)) ? 1 : 0;
            }
        }
        __syncthreads();
    }
    if (a < NA) order[(size_t)img * NA + rank] = a;
}

// ================= kernel 3: sequential greedy NMS (one block / image) =================
__global__ void yolo_nms_kernel(const float* __restrict__ preds,
                                const float* __restrict__ s,
                                const float* __restrict__ box,
                                const int*   __restrict__ lab,
                                const int*   __restrict__ order,
                                float* __restrict__ outF,
                                float* __restrict__ mc) {
    int img = blockIdx.x;
    int tid = threadIdx.x;
    __shared__ float kx1[MAX_DET], ky1[MAX_DET], kx2[MAX_DET], ky2[MAX_DET];
    __shared__ int   cnt, supp, clab, cidx;
    __shared__ float cb[4], cob[4], cs;

    float* ob  = outF + (size_t)img * MAX_DET * 4;                        // boxes  (8,100,4)
    float* os  = outF + (size_t)NB * MAX_DET * 4 + (size_t)img * MAX_DET; // scores (8,100,1)
    float* ol  = outF + (size_t)NB * MAX_DET * 5 + (size_t)img * MAX_DET; // labels (8,100,1)
    float* omc = mc   + (size_t)img * MPAD * NMC;

    for (int k = tid; k < MAX_DET * 4; k += blockDim.x) ob[k] = 0.0f;
    for (int k = tid; k < MAX_DET;     k += blockDim.x) { os[k] = 0.0f; ol[k] = 0.0f; }
    for (int k = tid; k < MPAD * NMC;  k += blockDim.x) omc[k] = 0.0f;
    if (tid == 0) cnt = 0;
    __syncthreads();

    for (int r = 0; r < NA; ++r) {
        if (tid == 0) {
            int i = order[(size_t)img * NA + r];
            cidx = i;
            cs   = s[(size_t)img * NA + i];
            clab = lab[(size_t)img * NA + i];
            float off = (float)clab * MAX_WH;
            #pragma unroll
            for (int c = 0; c < 4; ++c) {
                float b = box[((size_t)img * NA + i) * 4 + c];
                cb[c] = b; cob[c] = b + off;
            }
            supp = 0;
        }
        __syncthreads();
        if (cs < 0.0f) break;            // sorted: everything after is invalid

        if (tid < cnt) {                 // cnt <= 100 < blockDim
            float ix1 = fmaxf(kx1[tid], cob[0]);
            float iy1 = fmaxf(ky1[tid], cob[1]);
            float ix2 = fminf(kx2[tid], cob[2]);
            float iy2 = fminf(ky2[tid], cob[3]);
            float inter = fmaxf(ix2 - ix1, 0.0f) * fmaxf(iy2 - iy1, 0.0f);
            float a1 = (kx2[tid] - kx1[tid]) * (ky2[tid] - ky1[tid]);
            float a2 = (cob[2] - cob[0]) * (cob[3] - cob[1]);
            float iou = inter / (a1 + a2 - inter + 1e-7f);
            if (iou > IOU_THR) atomicExch(&supp, 1);
        }
        __syncthreads();

        if (!supp) {
            int k = cnt;                 // uniform read before update
            if (tid == 0) {
                kx1[k] = cob[0]; ky1[k] = cob[1]; kx2[k] = cob[2]; ky2[k] = cob[3];
                ob[k*4+0] = cb[0]; ob[k*4+1] = cb[1]; ob[k*4+2] = cb[2]; ob[k*4+3] = cb[3];
                os[k] = cs; ol[k] = (float)clab;
            }
            if (tid < NMC)
                omc[(size_t)k * NMC + tid] =
                    preds[(size_t)img * NCH * NA + (size_t)(4 + NCLS + tid) * NA + cidx];
            __syncthreads();
            if (tid == 0) cnt = k + 1;
        }
        __syncthreads();
        if (cnt >= MAX_DET) break;
    }
}

// ================= kernel 4: WMMA mask matmul + sigmoid + crop =================
// mc[MPAD x 32] @ proto[32 x 25600] via V_WMMA_F32_16X16X4_F32, wave32.
// A 16x4 f32 layout: lane<16 -> K=0,1 ; lane>=16 -> K=2,3 (ISA 7.12.2).
// C/D 16x16 f32: VGPR r -> M = r + 8*(lane>=16), N = lane&15.
// mc block is staged once per workgroup into LDS with the gfx1250 async
// global->LDS engine (ASYNCcnt + s_wait_asynccnt), then A-fragments come
// from ds_load.
__global__ void __launch_bounds__(128)
yolo_mask_wmma_kernel(const float* __restrict__ proto,
                      const float* __restrict__ mc,
                      const float* __restrict__ outF,
                      float* __restrict__ m160) {
    __shared__ float smc[MPAD * NMC];    // 14336 B of the 320KB WGP LDS

    int wave = blockIdx.x * (blockDim.x >> 5) + (threadIdx.x >> 5);
    int lane = threadIdx.x & 31;
    int nt   = wave % (PN / 16);
    int img  = wave / (PN / 16);         // all 4 waves of a block share one image

    const float* P  = proto + (size_t)img * NMC * PN;
    const float* MC = mc    + (size_t)img * MPAD * NMC;

    // ---- stage mc[112x32] into LDS (async DMA path when available) ----
#if defined(USE_ASYNC_LDS)
    for (int c = threadIdx.x; c < (MPAD * NMC) / 4; c += blockDim.x) {
        __builtin_amdgcn_global_load_async_to_lds_b128(
            (v4i_vs*)(MC + (size_t)c * 4),   // global source
            (v4i_vs*)&smc[c * 4],            // LDS destination
            /*imm offset*/0, /*cpol*/0);
    }
    __builtin_amdgcn_s_wait_asynccnt(0);
    __syncthreads();
#else
    for (int c = threadIdx.x; c < MPAD * NMC; c += blockDim.x) smc[c] = MC[c];
    __syncthreads();
#endif

    const int n0   = nt * 16;
    const int half = lane >> 4;     // 0: K 0/1, 1: K 2/3
    const int ln   = lane & 15;

    __builtin_prefetch(P + n0, 0, 1);   // global_prefetch_b8

    // B fragments (4x16 per k-step), reused for all 7 M tiles
    v2f bf[8];
    #pragma unroll
    for (int kk = 0; kk < 8; ++kk) {
        int kr = kk * 4 + half * 2;
        bf[kk].x = P[(size_t)kr       * PN + n0 + ln];
        bf[kk].y = P[(size_t)(kr + 1) * PN + n0 + ln];
    }

    const int n  = n0 + ln;
    const float fx = (float)(n % PW);
    const float fy = (float)(n / PW);

    #pragma unroll 1
    for (int mt = 0; mt < MPAD / 16; ++mt) {
        int m0 = mt * 16;
        v8f acc = {};
        #pragma unroll
        for (int kk = 0; kk < 8; ++kk) {
            int row = m0 + ln;
            int kc  = kk * 4 + half * 2;
            v2f a = *(const v2f*)&smc[row * NMC + kc];   // ds_load_b64
            acc = __builtin_amdgcn_wmma_f32_16x16x4_f32(
                /*neg_a=*/false, a, /*neg_b=*/false, bf[kk],
                /*c_mod=*/(short)0, acc, /*reuse_a=*/false, /*reuse_b=*/false);
        }
        #pragma unroll
        for (int r = 0; r < 8; ++r) {
            int M = m0 + r + half * 8;
            if (M < MAX_DET) {
                const float* bx = outF + ((size_t)img * MAX_DET + M) * 4;
                float x1 = bx[0] * 0.25f, y1 = bx[1] * 0.25f;  // 160/640 scale
                float x2 = bx[2] * 0.25f, y2 = bx[3] * 0.25f;
                bool inside = (fx >= x1) & (fx < x2) & (fy >= y1) & (fy < y2);
                float e = __expf(-acc[r]);
#if __has_builtin(__builtin_amdgcn_rcpf)
                float v = __builtin_amdgcn_rcpf(1.0f + e);     // v_rcp_f32
#else
                float v = 1.0f / (1.0f + e);
#endif
                m160[((size_t)img * MAX_DET + M) * PN + n] = inside ? v : 0.0f;
            }
        }
    }
}

// ================= kernel 5: bilinear 2x upsample + threshold =================
__global__ void yolo_resize_kernel(const float* __restrict__ m160,
                                   unsigned char* __restrict__ outM) {
    size_t g = (size_t)blockIdx.x * blockDim.x + threadIdx.x;
    const size_t total = (size_t)NB * MAX_DET * OH * OW;
    if (g >= total) return;
    int    ox = (int)(g % OW);
    int    oy = (int)((g / OW) % OH);
    size_t dm = g / ((size_t)OW * OH);      // img*MAX_DET + det

    float sx = (ox + 0.5f) * 0.5f - 0.5f;   // half-pixel centers, scale 0.5
    float sy = (oy + 0.5f) * 0.5f - 0.5f;
    sx = fminf(fmaxf(sx, 0.0f), (float)(PW - 1));
    sy = fminf(fmaxf(sy, 0.0f), (float)(PH - 1));
    int x0 = (int)sx, y0 = (int)sy;
    int x1 = (x0 + 1 < PW) ? x0 + 1 : PW - 1;
    int y1 = (y0 + 1 < PH) ? y0 + 1 : PH - 1;
    float wx = sx - (float)x0, wy = sy - (float)y0;

    const float* M = m160 + dm * PN;
    float v00 = M[(size_t)y0 * PW + x0], v01 = M[(size_t)y0 * PW + x1];
    float v10 = M[(size_t)y1 * PW + x0], v11 = M[(size_t)y1 * PW + x1];
    float v = v00 * (1.0f - wx) * (1.0f - wy) + v01 * wx * (1.0f - wy)
            + v10 * (1.0f - wx) * wy          + v11 * wx * wy;
    outM[g] = (v > 0.5f) ? (unsigned char)1 : (unsigned char)0;
}

// ================= launcher =================
extern "C" void kernel_launch(void* const* d_in, const int* in_sizes, int n_in,
                              void* d_out, int out_size, void* d_ws, size_t ws_size,
                              hipStream_t stream) {
    (void)in_sizes; (void)n_in; (void)out_size; (void)ws_size;
    const float* preds = (const float*)d_in[0];   // (8,116,8400) f32
    const float* proto = (const float*)d_in[1];   // (8,32,160,160) f32

    float* outF = (float*)d_out;                          // boxes, scores, labels (4800 f32)
    unsigned char* outM = (unsigned char*)(outF + (size_t)NB * MAX_DET * 6); // masks uint8

    float* ws   = (float*)d_ws;
    float* s    = ws + WS_S;
    float* box  = ws + WS_BOX;
    int*   lab  = (int*)(ws + WS_LAB);
    int*   ord  = (int*)(ws + WS_ORD);
    float* mc   = ws + WS_MC;
    float* m160 = ws + WS_M160;

    yolo_decode_kernel<<<(NB * NA + 255) / 256, 256, 0, stream>>>(preds, s, box, lab);

    dim3 rgrid((NA + 255) / 256, NB);
    yolo_rank_kernel<<<rgrid, 256, 0, stream>>>(s, ord);

    yolo_nms_kernel<<<NB, 128, 0, stream>>>(preds, s, box, lab, ord, outF, mc);

    // NB * (PN/16) = 12800 waves, 4 waves (128 threads) per block -> 3200 blocks
    yolo_mask_wmma_kernel<<<(NB * (PN / 16)) / 4, 128, 0, stream>>>(proto, mc, outF, m160);

    size_t total = (size_t)NB * MAX_DET * OH * OW;
    yolo_resize_kernel<<<(int)((total + 255) / 256), 256, 0, stream>>>(m160, outM);
}